// ConfidenceLossV2_69320772157832
// MI455X (gfx1250) — compile-verified
//
#include <hip/hip_runtime.h>
#include <hip/hip_bf16.h>

// Problem constants (from reference)
#define BB 8
#define CC 3
#define HH 512
#define WW 512
#define CF 64
#define HF 128
#define WF 128
#define GG 64
#define PLANE (HF * WF)        // 16384
#define IMG   (HH * WW)        // 262144 = 2^18

// Workspace layout (floats)
#define WS_RECOV 0            // sum(mse_recov * pos_mask)
#define WS_POSM  1            // sum(pos_mask)
#define WS_CNT   16           // counts[b*G+g]     (512)
#define WS_ERR   (16 + 512)   // sum_err[b*G+g]    (512)
#define WS_PCN   (16 + 1024)  // pos_cnt[b*G+g]    (512)
#define WS_TOTAL (16 + 1536)

typedef float v2f __attribute__((ext_vector_type(2)));
typedef float v8f __attribute__((ext_vector_type(8)));

__global__ void zero_ws_kernel(float* __restrict__ ws) {
    for (int i = threadIdx.x; i < WS_TOTAL; i += blockDim.x) ws[i] = 0.0f;
}

// ---------------------------------------------------------------------------
// Kernel 1: reco_error channel reduction via V_WMMA_F32_16X16X4_F32 + segment
// stats. Grid: (128, B). Block: 256 threads = 8 waves; each wave owns one
// 16-pixel tile (one row segment of the 128x128 plane) and runs 16 WMMA
// k-steps (4 channels each) with B = ones, so the matrix pipe performs the
// 64-deep channel reduction: C[m][n] = sum_k A[m,k].
// ---------------------------------------------------------------------------
__global__ void reco_stats_kernel(const float* __restrict__ enc1,
                                  const float* __restrict__ dec1,
                                  const float* __restrict__ masks,
                                  const int*   __restrict__ segs,
                                  float* __restrict__ ws) {
    __shared__ float lds_cnt[GG];
    __shared__ float lds_err[GG];
    __shared__ float lds_pos[GG];

    const int t = threadIdx.x;
    if (t < GG) { lds_cnt[t] = 0.0f; lds_err[t] = 0.0f; lds_pos[t] = 0.0f; }
    __syncthreads();

    const int b    = blockIdx.y;
    const int wave = t >> 5;
    const int lane = t & 31;

    const int tile  = blockIdx.x * 8 + wave;   // 0..1023 (tiles of 16 px)
    const int pb    = tile << 4;               // pixel base within plane
    const int h     = pb >> 7;                 // all 16 pixels share a row
    const int wbase = pb & 127;

    // A-matrix layout (16x4 f32): lanes 0-15 -> M=lane, K=0,1 ; lanes 16-31 -> M=lane-16, K=2,3
    const int m     = lane & 15;               // pixel (row of A) this lane feeds
    const int cpair = (lane >> 4) << 1;        // channel pair base: 0 or 2

    size_t base = (size_t)b * CF * PLANE + (size_t)cpair * PLANE
                + (size_t)h * WF + (size_t)(wbase + m);
    const float* ep = enc1 + base;
    const float* dp = dec1 + base;

    v8f acc = {};                              // 16x16 f32 accumulator
    v2f ones; ones.x = 1.0f; ones.y = 1.0f;    // B = all-ones (layout-free)

#pragma unroll
    for (int s = 0; s < 16; ++s) {             // 16 k-steps x 4 channels = 64
        float e0 = ep[0],      d0 = dp[0];
        float e1 = ep[PLANE],  d1 = dp[PLANE];
        float q0 = e0 - d0; q0 *= q0;
        float q1 = e1 - d1; q1 *= q1;
        v2f a; a.x = q0; a.y = q1;
        acc = __builtin_amdgcn_wmma_f32_16x16x4_f32(
                  false, a, false, ones, (short)0, acc, false, false);
        ep += 4 * PLANE;
        dp += 4 * PLANE;
    }

    // C layout: VGPR r -> lanes 0-15: M=r, lanes 16-31: M=r+8 (N = lane%16,
    // all columns identical since B==1). Lanes 0-7 take pixels 0-7, lanes
    // 16-23 take pixels 8-15.
    if ((lane & 15) < 8) {
        const int s = lane & 7;
        float v;
        switch (s) {
            case 0: v = acc[0]; break;
            case 1: v = acc[1]; break;
            case 2: v = acc[2]; break;
            case 3: v = acc[3]; break;
            case 4: v = acc[4]; break;
            case 5: v = acc[5]; break;
            case 6: v = acc[6]; break;
            default: v = acc[7]; break;
        }
        const int pix = s + ((lane >> 4) << 3);      // 0..15 within tile
        const int p   = pb + pix;
        const int hh  = p >> 7;
        const int wwp = p & 127;

        // strided subsample of segs / masks: [b, 0, hh*4, ww*4]
        const size_t soff = (size_t)b * IMG + (size_t)(hh << 2) * WW + (size_t)(wwp << 2);
        const int   g  = segs[soff] & (GG - 1);
        const float mk = masks[soff];
        const float pos = (mk < 0.5f && mk > 0.0f) ? 1.0f : 0.0f;
        const float err = v * (1.0f / 64.0f);        // mean over channels

        atomicAdd(&lds_cnt[g], 1.0f);
        atomicAdd(&lds_err[g], err);
        atomicAdd(&lds_pos[g], pos);
    }
    __syncthreads();

    if (t < GG) {
        atomicAdd(&ws[WS_CNT + b * GG + t], lds_cnt[t]);
        atomicAdd(&ws[WS_ERR + b * GG + t], lds_err[t]);
        atomicAdd(&ws[WS_PCN + b * GG + t], lds_pos[t]);
    }
}

// ---------------------------------------------------------------------------
// Kernel 2: masked reconstruction MSE over (B,3,H,W) -> two scalar sums.
// One thread per pixel; 3-channel inner loop; block tree-reduce; one f32
// atomic pair per block.
// ---------------------------------------------------------------------------
__global__ void recov_loss_kernel(const float* __restrict__ outputs,
                                  const float* __restrict__ inputs,
                                  const float* __restrict__ masks,
                                  float* __restrict__ ws) {
    __shared__ float s_mse[256];
    __shared__ float s_pos[256];

    const int i  = blockIdx.x * 256 + threadIdx.x;   // 0 .. B*H*W-1 (exact grid)
    const int b  = i >> 18;                          // / (H*W)
    const int hw = i & (IMG - 1);

    const float mk   = masks[i];
    const float posm = (mk > 0.0f) ? 1.0f : 0.0f;
    const bool  wall = (mk >= 0.5f);

    const size_t cb = (size_t)b * (CC * IMG) + (size_t)hw;
    float acc = 0.0f;
#pragma unroll
    for (int c = 0; c < CC; ++c) {
        const float tin = inputs[cb + (size_t)c * IMG];
        const float tgt = wall ? 0.0f : tin;
        const float d   = outputs[cb + (size_t)c * IMG] - tgt;
        acc += d * d;
    }

    s_mse[threadIdx.x] = acc * posm;
    s_pos[threadIdx.x] = posm;
    __syncthreads();
#pragma unroll
    for (int st = 128; st > 0; st >>= 1) {
        if (threadIdx.x < st) {
            s_mse[threadIdx.x] += s_mse[threadIdx.x + st];
            s_pos[threadIdx.x] += s_pos[threadIdx.x + st];
        }
        __syncthreads();
    }
    if (threadIdx.x == 0) {
        atomicAdd(&ws[WS_RECOV], s_mse[0]);
        atomicAdd(&ws[WS_POSM],  s_pos[0]);
    }
}

// ---------------------------------------------------------------------------
// Kernel 3: finalize. One block, 512 threads (= B*G entries).
// ---------------------------------------------------------------------------
__global__ void finalize_kernel(const float* __restrict__ ws,
                                float* __restrict__ out) {
    __shared__ float se[512];
    __shared__ float ss[512];

    const int t = threadIdx.x;                 // t = b*G + g
    const float cnt  = ws[WS_CNT + t];
    const float serr = ws[WS_ERR + t];
    const float pcnt = ws[WS_PCN + t];

    const bool  valid    = (cnt / (float)(HF * WF)) >= 0.01f;
    const float safe     = fmaxf(cnt, 1.0f);
    const float mean_err = serr / safe;
    const bool  pos_flag = valid && ((pcnt / safe) > 0.01f);
    const float sel      = pos_flag ? 1.0f : 0.0f;

    se[t] = mean_err * sel;
    ss[t] = sel;
    __syncthreads();
#pragma unroll
    for (int st = 256; st > 0; st >>= 1) {
        if (t < st) { se[t] += se[t + st]; ss[t] += ss[t + st]; }
        __syncthreads();
    }
    if (t == 0) {
        const float flat_pos_mean = se[0] / fmaxf(ss[0], 1.0f);
        const float loss_recov    = ws[WS_RECOV] / fmaxf(ws[WS_POSM], 1.0f);
        out[0] = loss_recov + flat_pos_mean;
    }
}

extern "C" void kernel_launch(void* const* d_in, const int* in_sizes, int n_in,
                              void* d_out, int out_size, void* d_ws, size_t ws_size,
                              hipStream_t stream) {
    const float* outputs = (const float*)d_in[0];
    const float* inputs  = (const float*)d_in[1];
    const float* enc1    = (const float*)d_in[2];
    const float* dec1    = (const float*)d_in[3];
    const float* masks   = (const float*)d_in[4];
    const int*   segs    = (const int*)d_in[5];
    float* out = (float*)d_out;
    float* ws  = (float*)d_ws;

    zero_ws_kernel<<<1, 256, 0, stream>>>(ws);

    // 128x128 plane per batch, 8 waves/block * 16 px/wave = 128 px/block
    reco_stats_kernel<<<dim3(128, BB), 256, 0, stream>>>(enc1, dec1, masks, segs, ws);

    // B*H*W = 2,097,152 pixels / 256 = 8192 blocks
    recov_loss_kernel<<<(BB * IMG) / 256, 256, 0, stream>>>(outputs, inputs, masks, ws);

    finalize_kernel<<<1, 512, 0, stream>>>(ws, out);
}